// Inception_62242666053645
// MI455X (gfx1250) — compile-verified
//
#include <hip/hip_runtime.h>

// ---------------------------------------------------------------------------
// Fused 3-branch causal dilated conv1d stack for MI455X (gfx1250).
// All intermediates live in LDS; every conv is an implicit GEMM on
// V_WMMA_F32_16X16X4_F32 (M=cout tile 16, N=16 positions, K=4 cin per issue).
// x tile is staged with GLOBAL_LOAD_ASYNC_TO_LDS_B32 (ASYNCcnt path).
// TILE_T=240 halves per-WG fixed staging/barrier cost vs the 112-wide tiling.
// ---------------------------------------------------------------------------

typedef __attribute__((ext_vector_type(2))) float v2f;
typedef __attribute__((ext_vector_type(8))) float v8f;

#define NTHR   256
#define NWAVE  8
#define TILE_T 240           // output positions per workgroup
#define LSTR   260           // LDS row stride (floats) for X / M1 / M2

// LDS float offsets (weights zero-padded to (K, coPad, cinPad))
#define W7_1_OFF  0          // 14x8x3  -> K3 coPad16 cinPad8  : 384
#define W10_1_OFF 384
#define W15_1_OFF 768
#define W7_3_OFF  1152       // 32x14x3 -> K3 coPad32 cinPad16 : 1536
#define W10_3_OFF 2688       // 21x14x3 -> same padded box
#define W15_3_OFF 4224
#define W10_5_OFF 5760       // 32x21x2 -> K2 coPad32 cinPad24 : 1536
#define W15_5_OFF 7296       // 32x21x3 -> K3 coPad32 cinPad24 : 2304
#define WTOT      9600
#define X_OFF     9600       //  8 x 260
#define M1_OFF    11680      // 16 x 260
#define M2_OFF    15840      // 24 x 260
#define LDS_FLOATS 22080     // 88320 bytes (CDNA5 allows up to 320KB/WG)

#define L_IN  16384
#define L_OUT 16378
#define C_IN  8

// Single-VALU-intent ReLU.
__device__ __forceinline__ float relu1(float x) {
  return __builtin_amdgcn_fmed3f(x, 0.0f, __builtin_inff());
}

// Copy one weight tensor (O,C,K row-major) into the padded LDS layout:
// dst[(k*coPad + o)*cinPad + c]
__device__ __forceinline__ void stage_w(float* dst, const float* __restrict__ w,
                                        int O, int C, int K, int coPad, int cinPad,
                                        int tid) {
  const int tot = O * C * K;
  for (int i = tid; i < tot; i += NTHR) {
    const int k = i % K;
    const int c = (i / K) % C;
    const int o = i / (K * C);
    dst[(k * coPad + o) * cinPad + c] = w[i];
  }
}

// One 16(cout) x 16(pos) output tile of a dilated conv as WMMA GEMM.
//  A (16x4): lanes 0-15 -> rows M, K=0,1 in vgpr0/1; lanes 16-31 -> K=2,3.
//  B (4x16): lanes 0-15 -> cols N with rows 0,1; lanes 16-31 -> rows 2,3.
template<int CINPAD, int COPAD, int K>
__device__ __forceinline__ v8f conv_tile(const float* __restrict__ Wl,
                                         const float* __restrict__ S,
                                         int coBase, int p0, int dil, int lane) {
  const int m  = lane & 15;        // row of A == col of B/D for this lane
  const int hi = lane >> 4;
  v8f acc = {0.f, 0.f, 0.f, 0.f, 0.f, 0.f, 0.f, 0.f};
#pragma unroll
  for (int k = 0; k < K; ++k) {
    const int sp = p0 + k * dil + m;
#pragma unroll
    for (int c4 = 0; c4 < CINPAD / 4; ++c4) {
      const int cin = c4 * 4 + hi * 2;
      v2f a, b;
      const float* wr = &Wl[(k * COPAD + coBase + m) * CINPAD + cin];
      a.x = wr[0];
      a.y = wr[1];
      b.x = S[cin * LSTR + sp];
      b.y = S[(cin + 1) * LSTR + sp];
      acc = __builtin_amdgcn_wmma_f32_16x16x4_f32(false, a, false, b, (short)0,
                                                  acc, false, false);
    }
  }
  return acc;
}

// Stage 1: 8ch -> 14ch, K=3 d=1, ReLU, into M1 (rows 14..15 forced to 0).
// 16 tiles over [0, 256) -> exactly two per wave, unrolled.
__device__ __forceinline__ void stage1_conv(const float* __restrict__ Wl,
                                            const float* __restrict__ Xb,
                                            float* __restrict__ M1b,
                                            int wave, int lane) {
  const int n = lane & 15, hi = lane >> 4;
#pragma unroll
  for (int iter = 0; iter < 2; ++iter) {
    const int p0 = (wave + iter * NWAVE) * 16;
    v8f acc = conv_tile<8, 16, 3>(Wl, Xb, 0, p0, 1, lane);
#pragma unroll
    for (int v = 0; v < 8; ++v) {
      const int co = hi * 8 + v;
      M1b[co * LSTR + p0 + n] = (co < 14) ? relu1(acc[v]) : 0.f;
    }
  }
}

// Stage 2 (branches 10/15): 14ch -> 21ch, K=3 d=2, ReLU, into M2 over
// positions [t0-8, t0+T+8). Causal left-pad: s < 0 rows are zeroed.
// 32 tasks (16 tiles x 2 cout tiles) -> exactly four per wave, unrolled.
__device__ __forceinline__ void stage2_mid(const float* __restrict__ Wl,
                                           const float* __restrict__ M1b,
                                           float* __restrict__ M2b,
                                           int t0, int wave, int lane) {
  const int n = lane & 15, hi = lane >> 4;
#pragma unroll
  for (int iter = 0; iter < 4; ++iter) {
    const int task = wave + iter * NWAVE;
    const int tile = task >> 1, coT = (task & 1) * 16;
    const int p0 = tile * 16;
    v8f acc = conv_tile<16, 32, 3>(Wl, M1b, coT, p0, 2, lane);
    const int s = t0 - 8 + p0 + n;   // absolute m2 position
#pragma unroll
    for (int v = 0; v < 8; ++v) {
      const int co = coT + hi * 8 + v;
      if (co < 24)
        M2b[co * LSTR + p0 + n] = (co < 21 && s >= 0) ? relu1(acc[v]) : 0.f;
    }
  }
}

// Final 32ch conv of a branch -> ReLU -> global store at channel chOff.
// 30 tasks (15 position tiles x 2 cout tiles).
template<int CINPAD, int K>
__device__ __forceinline__ void final_stage(const float* __restrict__ Wl,
                                            const float* __restrict__ S,
                                            float* __restrict__ out,
                                            int bIdx, int t0, int chOff,
                                            int srcBase, int dil,
                                            int wave, int lane) {
  const int n = lane & 15, hi = lane >> 4;
  for (int task = wave; task < 30; task += NWAVE) {
    const int tile = task >> 1, coT = (task & 1) * 16;
    const int p0 = tile * 16;
    v8f acc = conv_tile<CINPAD, 32, K>(Wl, S, coT, srcBase + p0, dil, lane);
    const int t = t0 + p0 + n;
    if (t < L_OUT) {
#pragma unroll
      for (int v = 0; v < 8; ++v) {
        const int co = coT + hi * 8 + v;
        out[((size_t)bIdx * 96 + chOff + co) * L_OUT + t] = relu1(acc[v]);
      }
    }
  }
}

__global__ __launch_bounds__(NTHR) void incept_fused(
    const float* __restrict__ x,
    const float* __restrict__ w7_1,  const float* __restrict__ w7_3,
    const float* __restrict__ w10_1, const float* __restrict__ w10_3,
    const float* __restrict__ w10_5,
    const float* __restrict__ w15_1, const float* __restrict__ w15_3,
    const float* __restrict__ w15_5,
    float* __restrict__ out) {
  __shared__ float lds[LDS_FLOATS];
  const int tid  = threadIdx.x;
  const int lane = tid & 31;
  const int wave = tid >> 5;
  const int bIdx = blockIdx.y;
  const int t0   = blockIdx.x * TILE_T;

  // ---- stage x tile asynchronously: global -> LDS without VGPR round-trip.
  // Local p maps to absolute position t0-8+p, clamped at both edges.
  // GVS form: mem = SGPR64(base) + VGPR32(offset); LDS addr = VGPR32(dst).
  {
    const float* xb = x + (size_t)bIdx * C_IN * L_IN;   // uniform -> SGPR pair
    for (int i = tid; i < C_IN * LSTR; i += NTHR) {
      const int c = i / LSTR, p = i % LSTR;
      int g = t0 - 8 + p;
      g = g < 0 ? 0 : (g > L_IN - 1 ? L_IN - 1 : g);
      unsigned ldsOff = (unsigned)(X_OFF + c * LSTR + p) * 4u;  // lds[] is at
      unsigned gOff   = (unsigned)(c * L_IN + g) * 4u;          // alloc offset 0
      asm volatile("global_load_async_to_lds_b32 %0, %1, %2"
                   :
                   : "v"(ldsOff), "v"(gOff), "s"(xb)
                   : "memory");
    }
  }

  // ---- stage weights (zero-padded) while the async x copy is in flight ----
  for (int i = tid; i < WTOT; i += NTHR) lds[i] = 0.f;
  __syncthreads();
  stage_w(&lds[W7_1_OFF],  w7_1,  14,  8, 3, 16,  8, tid);
  stage_w(&lds[W10_1_OFF], w10_1, 14,  8, 3, 16,  8, tid);
  stage_w(&lds[W15_1_OFF], w15_1, 14,  8, 3, 16,  8, tid);
  stage_w(&lds[W7_3_OFF],  w7_3,  32, 14, 3, 32, 16, tid);
  stage_w(&lds[W10_3_OFF], w10_3, 21, 14, 3, 32, 16, tid);
  stage_w(&lds[W15_3_OFF], w15_3, 21, 14, 3, 32, 16, tid);
  stage_w(&lds[W10_5_OFF], w10_5, 32, 21, 2, 32, 24, tid);
  stage_w(&lds[W15_5_OFF], w15_5, 32, 21, 3, 32, 24, tid);

  asm volatile("s_wait_asynccnt 0x0" ::: "memory");  // x tile landed in LDS
  __syncthreads();

  const float* Xb  = &lds[X_OFF];
  float* M1b = &lds[M1_OFF];
  float* M2b = &lds[M2_OFF];

  // ---- branch 7: conv(3,d1) -> conv(3,d2) -> out[0..32) ----
  stage1_conv(&lds[W7_1_OFF], Xb, M1b, wave, lane);
  __syncthreads();
  final_stage<16, 3>(&lds[W7_3_OFF], M1b, out, bIdx, t0, 0, /*srcBase=*/8,
                     /*dil=*/2, wave, lane);
  __syncthreads();

  // ---- branch 10: conv(3,d1) -> conv(3,d2) -> conv(2,d4,pad4) -> out[32..64) ----
  stage1_conv(&lds[W10_1_OFF], Xb, M1b, wave, lane);
  __syncthreads();
  stage2_mid(&lds[W10_3_OFF], M1b, M2b, t0, wave, lane);
  __syncthreads();
  // m3[t] = w0*m2[t-4] + w1*m2[t] -> src local base 4, dil 4
  final_stage<24, 2>(&lds[W10_5_OFF], M2b, out, bIdx, t0, 32, /*srcBase=*/4,
                     /*dil=*/4, wave, lane);
  __syncthreads();

  // ---- branch 15: conv(3,d1) -> conv(3,d2) -> conv(3,d4,pad8) -> out[64..96) ----
  stage1_conv(&lds[W15_1_OFF], Xb, M1b, wave, lane);
  __syncthreads();
  stage2_mid(&lds[W15_3_OFF], M1b, M2b, t0, wave, lane);
  __syncthreads();
  // m3[t] = w0*m2[t-8] + w1*m2[t-4] + w2*m2[t] -> src local base 0, dil 4
  final_stage<24, 3>(&lds[W15_5_OFF], M2b, out, bIdx, t0, 64, /*srcBase=*/0,
                     /*dil=*/4, wave, lane);
}

extern "C" void kernel_launch(void* const* d_in, const int* in_sizes, int n_in,
                              void* d_out, int out_size, void* d_ws, size_t ws_size,
                              hipStream_t stream) {
  (void)in_sizes; (void)n_in; (void)out_size; (void)d_ws; (void)ws_size;
  const float* x     = (const float*)d_in[0];
  const float* w7_1  = (const float*)d_in[1];
  const float* w7_3  = (const float*)d_in[2];
  const float* w10_1 = (const float*)d_in[3];
  const float* w10_3 = (const float*)d_in[4];
  const float* w10_5 = (const float*)d_in[5];
  const float* w15_1 = (const float*)d_in[6];
  const float* w15_3 = (const float*)d_in[7];
  const float* w15_5 = (const float*)d_in[8];
  float* out = (float*)d_out;

  dim3 grid((L_OUT + TILE_T - 1) / TILE_T, 64);  // 69 x 64 workgroups
  incept_fused<<<grid, NTHR, 0, stream>>>(x, w7_1, w7_3, w10_1, w10_3, w10_5,
                                          w15_1, w15_3, w15_5, out);
}